// ResidualModule_25640954757916
// MI455X (gfx1250) — compile-verified
//
#include <hip/hip_runtime.h>

typedef __attribute__((ext_vector_type(2))) float v2f;
typedef __attribute__((ext_vector_type(8))) float v8f;

#define C 128

// ---------------------------------------------------------------------------
// Scatter-add: one wave (32 lanes) per edge, each lane handles 4 channels.
// Source row read is a fully coalesced 512B burst; destination updates are
// hardware f32 atomics that resolve in L2 (all agg buffers fit in 192MB L2).
// ---------------------------------------------------------------------------
__global__ __launch_bounds__(256) void scatter_add_kernel(
    const float* __restrict__ src_feat,
    const int* __restrict__ edges,   // [2, E] row-major: row0 = src, row1 = dst
    float* __restrict__ agg, int E)
{
    int warp = (int)((blockIdx.x * blockDim.x + threadIdx.x) >> 5);
    int lane = threadIdx.x & 31;
    if (warp >= E) return;
    int s = edges[warp];        // source node
    int t = edges[E + warp];    // target node
    const float4 v = *(const float4*)(src_feat + (size_t)s * C + lane * 4);
    float* dstp = agg + (size_t)t * C + lane * 4;
    unsafeAtomicAdd(dstp + 0, v.x);
    unsafeAtomicAdd(dstp + 1, v.y);
    unsafeAtomicAdd(dstp + 2, v.z);
    unsafeAtomicAdd(dstp + 3, v.w);
}

// ---------------------------------------------------------------------------
// Fused 3-term GEMM:  out = act( A0@W0 + A1@W1 + A2@W2 (+ resid) )
// M x 128 @ 128 x 128, f32.  Block = 16 rows x 128 cols, 8 waves, each wave
// owns one 16x16 tile and runs 3 x 32 V_WMMA_F32_16X16X4_F32 accumulations.
//
// Operand packing per CDNA5 ISA 7.12.2 (wave32):
//   A 16x4:  lane<16 -> K={k0,k0+1}, lane>=16 -> K={k0+2,k0+3}; M = lane%16
//   B 4x16:  same half/K split; N = lane%16
//   C/D:     vgpr v, lane l -> M = v + 8*(l/16), N = l%16
// LDS A-tile rows padded to 132 floats -> column reads hit distinct banks.
// ---------------------------------------------------------------------------
__global__ __launch_bounds__(256) void gemm3_wmma_kernel(
    const float* __restrict__ A0, const float* __restrict__ W0,
    const float* __restrict__ A1, const float* __restrict__ W1,
    const float* __restrict__ A2, const float* __restrict__ W2,
    const float* __restrict__ resid,   // may be nullptr
    float* __restrict__ out, int M)
{
    __shared__ float lA[3][16][132];

    const int row0 = blockIdx.x * 16;
    const float* As[3] = {A0, A1, A2};

    // Cooperative stage of three 16x128 A tiles (512 float4 each).
    #pragma unroll
    for (int t = 0; t < 3; ++t) {
        for (int i = threadIdx.x; i < 512; i += 256) {
            int r  = i >> 5;
            int c4 = (i & 31) << 2;
            int gr = row0 + r;
            if (gr >= M) gr = M - 1;            // safe duplicate for tail
            *(float4*)&lA[t][r][c4] =
                *(const float4*)(As[t] + (size_t)gr * C + c4);
        }
    }
    __syncthreads();

    const int wave = threadIdx.x >> 5;      // 0..7 -> column tile
    const int lane = threadIdx.x & 31;
    const int half = lane >> 4;             // 0: K group 0, 1: K group +2
    const int lr   = lane & 15;
    const int col  = wave * 16 + lr;
    const int koff = 2 * half;

    v8f acc = {};
    #pragma unroll 8
    for (int k0 = 0; k0 < C; k0 += 4) {
        const int ka = k0 + koff;

        v2f a0; a0.x = lA[0][lr][ka]; a0.y = lA[0][lr][ka + 1];
        v2f b0; b0.x = W0[ka * C + col]; b0.y = W0[(ka + 1) * C + col];
        acc = __builtin_amdgcn_wmma_f32_16x16x4_f32(
                  false, a0, false, b0, (short)0, acc, false, false);

        v2f a1; a1.x = lA[1][lr][ka]; a1.y = lA[1][lr][ka + 1];
        v2f b1; b1.x = W1[ka * C + col]; b1.y = W1[(ka + 1) * C + col];
        acc = __builtin_amdgcn_wmma_f32_16x16x4_f32(
                  false, a1, false, b1, (short)0, acc, false, false);

        v2f a2; a2.x = lA[2][lr][ka]; a2.y = lA[2][lr][ka + 1];
        v2f b2; b2.x = W2[ka * C + col]; b2.y = W2[(ka + 1) * C + col];
        acc = __builtin_amdgcn_wmma_f32_16x16x4_f32(
                  false, a2, false, b2, (short)0, acc, false, false);
    }

    // Epilogue: optional residual, ReLU, store.
    #pragma unroll
    for (int v = 0; v < 8; ++v) {
        int r  = row0 + v + 8 * half;
        if (r >= M) continue;
        float val = acc[v];
        if (resid) val += resid[(size_t)r * C + col];
        val = fmaxf(val, 0.0f);
        out[(size_t)r * C + col] = val;
    }
}

// ---------------------------------------------------------------------------
// Host-side orchestration (graph-capture safe: only async ops on `stream`).
// ---------------------------------------------------------------------------
extern "C" void kernel_launch(void* const* d_in, const int* in_sizes, int n_in,
                              void* d_out, int out_size, void* d_ws, size_t ws_size,
                              hipStream_t stream)
{
    const float* hd    = (const float*)d_in[0];
    const float* hp    = (const float*)d_in[1];
    const int*   e_dd  = (const int*)d_in[2];
    const int*   e_dp  = (const int*)d_in[3];
    const int*   e_pd  = (const int*)d_in[4];
    const int*   e_pp  = (const int*)d_in[5];
    const float* w1_sd = (const float*)d_in[6];
    const float* w1_sp = (const float*)d_in[7];
    const float* w1_dd = (const float*)d_in[8];
    const float* w1_dp = (const float*)d_in[9];
    const float* w1_pd = (const float*)d_in[10];
    const float* w1_pp = (const float*)d_in[11];
    const float* w2_sd = (const float*)d_in[12];
    const float* w2_sp = (const float*)d_in[13];
    const float* w2_dd = (const float*)d_in[14];
    const float* w2_dp = (const float*)d_in[15];
    const float* w2_pd = (const float*)d_in[16];
    const float* w2_pp = (const float*)d_in[17];

    const int ND = in_sizes[0] / C;      // 50000
    const int NP = in_sizes[1] / C;      // 50000
    const int E  = in_sizes[2] / 2;      // 640000

    const size_t segD = (size_t)ND * C;
    const size_t segP = (size_t)NP * C;

    float* ws     = (float*)d_ws;
    float* agg_dd = ws;                       // ND x C
    float* agg_pd = agg_dd + segD;            // ND x C
    float* agg_dp = agg_pd + segD;            // NP x C
    float* agg_pp = agg_dp + segP;            // NP x C
    float* od     = agg_pp + segP;            // ND x C (layer-1 drug act)
    float* op     = od + segD;                // NP x C (layer-1 prot act)

    float* outd = (float*)d_out;
    float* outp = outd + segD;

    const dim3 sblk(256);
    const dim3 sgrd((unsigned)((E + 7) / 8));     // 8 waves/block, 1 wave/edge
    const dim3 gblk(256);
    const dim3 ggrdD((unsigned)((ND + 15) / 16));
    const dim3 ggrdP((unsigned)((NP + 15) / 16));

    // ---------------- Layer 1 ----------------
    hipMemsetAsync(agg_dd, 0, (2 * segD + 2 * segP) * sizeof(float), stream);
    scatter_add_kernel<<<sgrd, sblk, 0, stream>>>(hd, e_dd, agg_dd, E);
    scatter_add_kernel<<<sgrd, sblk, 0, stream>>>(hp, e_pd, agg_pd, E);
    scatter_add_kernel<<<sgrd, sblk, 0, stream>>>(hd, e_dp, agg_dp, E);
    scatter_add_kernel<<<sgrd, sblk, 0, stream>>>(hp, e_pp, agg_pp, E);

    gemm3_wmma_kernel<<<ggrdD, gblk, 0, stream>>>(
        hd, w1_sd, agg_dd, w1_dd, agg_pd, w1_pd, nullptr, od, ND);
    gemm3_wmma_kernel<<<ggrdP, gblk, 0, stream>>>(
        hp, w1_sp, agg_pp, w1_pp, agg_dp, w1_dp, nullptr, op, NP);

    // ---------------- Layer 2 ----------------
    hipMemsetAsync(agg_dd, 0, (2 * segD + 2 * segP) * sizeof(float), stream);
    scatter_add_kernel<<<sgrd, sblk, 0, stream>>>(od, e_dd, agg_dd, E);
    scatter_add_kernel<<<sgrd, sblk, 0, stream>>>(op, e_pd, agg_pd, E);
    scatter_add_kernel<<<sgrd, sblk, 0, stream>>>(od, e_dp, agg_dp, E);
    scatter_add_kernel<<<sgrd, sblk, 0, stream>>>(op, e_pp, agg_pp, E);

    gemm3_wmma_kernel<<<ggrdD, gblk, 0, stream>>>(
        od, w2_sd, agg_dd, w2_dd, agg_pd, w2_pd, hd, outd, ND);
    gemm3_wmma_kernel<<<ggrdP, gblk, 0, stream>>>(
        op, w2_sp, agg_pp, w2_pp, agg_dp, w2_dp, hp, outp, NP);
}